// Sampler_65919158059159
// MI455X (gfx1250) — compile-verified
//
#include <hip/hip_runtime.h>
#include <math.h>
#include <stdint.h>

#define BLOCK 512
#define NB1   2048   // bits [31:21]
#define NB2   2048   // bits [20:10]
#define NB3   1024   // bits [9:0]
#define TILE  (4 * BLOCK)   // 2048 floats = 8KB per staged tile

// ---------- CDNA5 feature probes ----------
#if defined(__has_builtin)
#  if __has_builtin(__builtin_amdgcn_global_load_async_to_lds_b128)
#    define HAVE_ASYNC_CP 1
#  endif
#  if __has_builtin(__builtin_amdgcn_tensor_load_to_lds) && \
      __has_builtin(__builtin_amdgcn_s_wait_tensorcnt)
#    define HAVE_TDM 1
#  endif
#endif
#ifndef HAVE_ASYNC_CP
#  define HAVE_ASYNC_CP 0
#endif
#ifndef HAVE_TDM
#  define HAVE_TDM 0
#endif

#if defined(__has_builtin) && __has_builtin(__builtin_amdgcn_s_wait_asynccnt)
#  define WAIT_ASYNC(n) __builtin_amdgcn_s_wait_asynccnt(n)
#else
#  define WAIT_ASYNC(n) asm volatile("s_wait_asynccnt %0" :: "i"(n) : "memory")
#endif

#if HAVE_ASYNC_CP
// Param types from the round-1 diagnostic: v4i AS(1)* , lds side AS(3)*.
typedef __attribute__((__vector_size__(4 * sizeof(int)))) int v4i;
typedef __attribute__((address_space(1))) v4i g_v4i;
typedef __attribute__((address_space(3))) v4i l_v4i;
__device__ __forceinline__ void async_cp16(const float* g, float* l) {
  __builtin_amdgcn_global_load_async_to_lds_b128((g_v4i*)g, (l_v4i*)l, 0, 0);
}
#endif

#if HAVE_TDM
typedef unsigned int u32x4 __attribute__((ext_vector_type(4)));
typedef int          i32x8 __attribute__((ext_vector_type(8)));
typedef int          i32x4 __attribute__((ext_vector_type(4)));
#define WAIT_TENSOR(n) __builtin_amdgcn_s_wait_tensorcnt(n)
// Tensor DMA descriptor per CDNA5 ISA:
// 1-D tile of `elems` 4-byte elements, global -> LDS, no multicast/pad/iterate.
// Toolchain uses the 6-arg builtin form: (g0, g1, g2, g3, g4, cpol).
__device__ __forceinline__ void tdm_load_tile(const float* gsrc, float* ldst, int elems) {
  unsigned long long ga = (unsigned long long)(uintptr_t)gsrc;
  unsigned la = (unsigned)(size_t)(__attribute__((address_space(3))) const char*)(const void*)ldst;
  u32x4 g0;
  g0.x = 0x1u;                                               // count=1 (valid user D#)
  g0.y = la;                                                 // lds_addr (bytes)
  g0.z = (unsigned)(ga & 0xffffffffu);                       // global_addr[31:0]
  g0.w = (unsigned)((ga >> 32) & 0x1ffffffu) | (2u << 30);   // global_addr[56:32] | type=2
  i32x8 g1;
  g1[0] = (int)(2u << 16);                                   // wg_mask=0, data_size=2 (4B)
  g1[1] = (int)(((unsigned)elems & 0xffffu) << 16);          // tensor_dim0[15:0] @ [63:48]
  g1[2] = (int)((((unsigned)elems >> 16) & 0xffffu) | (1u << 16)); // dim0 hi | tensor_dim1=1
  g1[3] = (int)(((unsigned)elems & 0xffffu) << 16);          // tile_dim0 @ [127:112]
  g1[4] = 0;                                                 // tile_dim1=0, tile_dim2=0
  g1[5] = (int)(unsigned)elems;                              // tensor_dim0_stride lo32
  g1[6] = 0;                                                 // stride hi, dim1_stride lo
  g1[7] = 0;
  i32x4 z4 = {0, 0, 0, 0};                                   // groups 2/3 unused (<=2D)
  i32x8 z8 = {0, 0, 0, 0, 0, 0, 0, 0};                       // extra group in 6-arg form
  __builtin_amdgcn_tensor_load_to_lds(g0, g1, z4, z4, z8, 0);
}
#endif

// order-preserving float -> uint mapping (larger float <=> larger uint)
__device__ __forceinline__ unsigned ford(float f) {
  unsigned u = __float_as_uint(f);
  return (u & 0x80000000u) ? ~u : (u | 0x80000000u);
}
// exact inverse of ford()
__device__ __forceinline__ float iford(unsigned u) {
  unsigned b = (u & 0x80000000u) ? (u & 0x7fffffffu) : ~u;
  return __uint_as_float(b);
}

enum { MODE_RANK = 0, MODE_MASS = 1, MODE_MASS_FINAL = 2 };

// Block-wide suffix scan over histogram h[0..nb-1] (ascending bin value),
// then locate the radix-descent bucket. Whole block must call this.
__device__ void scan_detect(float* h, float* part, int nb, int mode,
                            float KR, bool scaleByTotal,
                            int* s_bin, float* s_rem) {
  const int tid  = threadIdx.x;
  const int CH   = nb / BLOCK;      // 4 or 2
  const int base = tid * CH;
  float ls[4];
  float tot = 0.f;
  for (int c = CH - 1; c >= 0; --c) { tot += h[base + c]; ls[c] = tot; }
  if (tid == 0) { *s_bin = 0x7fffffff; *s_rem = 0.f; }
  part[tid] = tot;
  __syncthreads();
  for (int off = 1; off < BLOCK; off <<= 1) {   // Hillis-Steele suffix scan
    float add = (tid + off < BLOCK) ? part[tid + off] : 0.f;
    __syncthreads();
    part[tid] += add;
    __syncthreads();
  }
  const float chunkExcl = part[tid] - tot;
  const float R = scaleByTotal ? KR * part[0] : KR;  // part[0] == total mass (Z)
  for (int c = 0; c < CH; ++c) {
    int   b   = base + c;
    float S   = ls[c] + chunkExcl;
    float hb  = h[b];
    float Sex = S - hb;
    bool hit;
    if (mode == MODE_RANK)      hit = (S >= R) && (Sex < R);
    else if (mode == MODE_MASS) hit = (Sex <= R) && (S > R);
    else                        hit = (hb > 0.f) && (Sex <= R);
    if (hit) atomicMin(s_bin, b);
  }
  __syncthreads();
  if (*s_bin == 0x7fffffff) {       // defensive fallback: lowest nonempty bin
    for (int c = 0; c < CH; ++c) {
      int b = base + c;
      if (h[b] > 0.f) atomicMin(s_bin, b);
    }
    __syncthreads();
  }
  int win = *s_bin;
  if (win >= base && win < base + CH) {
    int c = win - base;
    float Sex = (ls[c] + chunkExcl) - h[win];
    float rem = R - Sex;
    *s_rem = rem > 0.f ? rem : 0.f;
  }
  __syncthreads();
}

template <bool USE_WS>
__global__ __launch_bounds__(BLOCK) void sampler_kernel(
    const float* __restrict__ logits,
    const float* __restrict__ temps,
    const float* __restrict__ topps,
    const int*   __restrict__ topks,
    const float* __restrict__ noise,
    float* __restrict__ out,
    unsigned* __restrict__ uws, int v)
{
  const int row = blockIdx.x;
  const int tid = threadIdx.x;
  const float* xrow = logits + (size_t)row * v;
  const float* nrow = noise  + (size_t)row * v;
  unsigned* urow = USE_WS ? (uws + (size_t)row * v) : nullptr;

  __shared__ float h[NB1];
  __shared__ float part[BLOCK];
  __shared__ float redv[BLOCK];
  __shared__ float redv2[BLOCK];
  __shared__ int   redi[BLOCK];
  __shared__ int   s_bin;
  __shared__ float s_rem;
#if HAVE_TDM || HAVE_ASYNC_CP
  __shared__ float stage[2][TILE];
#endif

  const float t      = temps[row];
  const bool  greedy = (t <= 1e-10f);
  const float st     = greedy ? 1.0f : t;     // reference's safe_t
  const float p      = topps[row];
  int k = topks[row];
  if (k > v) k = v;

  // ---------------- Pass 1: level-1 count histogram + max / argmax / 2nd-max
  for (int b = tid; b < NB1; b += BLOCK) h[b] = 0.f;
  __syncthreads();

  float m1 = -INFINITY, m2 = -INFINITY;
  int   i1 = 0;

#if HAVE_TDM
  {
    const int  nfull = v / TILE;
    const bool wave0 = (tid < 32);        // wave-uniform; TDM is a per-wave DMA
    if (wave0 && nfull > 0) tdm_load_tile(xrow, &stage[0][0], TILE);
    for (int tile = 0; tile < nfull; ++tile) {
      const int buf = tile & 1;
      if (wave0) {
        if (tile + 1 < nfull) {
          tdm_load_tile(xrow + (size_t)(tile + 1) * TILE, &stage[buf ^ 1][0], TILE);
          WAIT_TENSOR(1);                 // current tile done, next in flight
        } else {
          WAIT_TENSOR(0);
        }
      }
      __syncthreads();
      const int ibase = tile * TILE + tid * 4;
      for (int c = 0; c < 4; ++c) {
        float xv = stage[buf][tid * 4 + c] / st;
        unsigned u = ford(xv);
        if (USE_WS) urow[ibase + c] = u;
        atomicAdd(&h[u >> 21], 1.0f);
        if (xv > m1) { m2 = m1; m1 = xv; i1 = ibase + c; }
        else if (xv > m2) m2 = xv;
      }
      __syncthreads();                    // buffer reuse next iteration
    }
    for (int i = nfull * TILE + tid; i < v; i += BLOCK) {   // tail
      float xv = xrow[i] / st;
      unsigned u = ford(xv);
      if (USE_WS) urow[i] = u;
      atomicAdd(&h[u >> 21], 1.0f);
      if (xv > m1) { m2 = m1; m1 = xv; i1 = i; }
      else if (xv > m2) m2 = xv;
    }
  }
#elif HAVE_ASYNC_CP
  {
    const int nfull = v / TILE;
    if (nfull > 0) async_cp16(xrow + tid * 4, &stage[0][tid * 4]);
    for (int tile = 0; tile < nfull; ++tile) {
      const int buf = tile & 1;
      if (tile + 1 < nfull) {
        async_cp16(xrow + (size_t)(tile + 1) * TILE + tid * 4, &stage[buf ^ 1][tid * 4]);
        WAIT_ASYNC(1);
      } else {
        WAIT_ASYNC(0);
      }
      __syncthreads();
      const int ibase = tile * TILE + tid * 4;
      for (int c = 0; c < 4; ++c) {
        float xv = stage[buf][tid * 4 + c] / st;
        unsigned u = ford(xv);
        if (USE_WS) urow[ibase + c] = u;
        atomicAdd(&h[u >> 21], 1.0f);
        if (xv > m1) { m2 = m1; m1 = xv; i1 = ibase + c; }
        else if (xv > m2) m2 = xv;
      }
      __syncthreads();
    }
    for (int i = nfull * TILE + tid; i < v; i += BLOCK) {
      float xv = xrow[i] / st;
      unsigned u = ford(xv);
      if (USE_WS) urow[i] = u;
      atomicAdd(&h[u >> 21], 1.0f);
      if (xv > m1) { m2 = m1; m1 = xv; i1 = i; }
      else if (xv > m2) m2 = xv;
    }
  }
#else
  for (int i = tid; i < v; i += BLOCK) {
    float xv = xrow[i] / st;
    unsigned u = ford(xv);
    if (USE_WS) urow[i] = u;
    atomicAdd(&h[u >> 21], 1.0f);
    if (xv > m1) { m2 = m1; m1 = xv; i1 = i; }
    else if (xv > m2) m2 = xv;
  }
#endif
  __syncthreads();

  // block reduce: (max, first-index argmax, second-max)
  redv[tid] = m1; redi[tid] = i1; redv2[tid] = m2;
  for (int s = BLOCK / 2; s > 0; s >>= 1) {
    __syncthreads();
    if (tid < s) {
      float b1 = redv[tid + s]; int bi = redi[tid + s]; float b2 = redv2[tid + s];
      float a1 = redv[tid];     int ai = redi[tid];     float a2 = redv2[tid];
      if (b1 > a1 || (b1 == a1 && bi < ai)) {
        redv[tid] = b1; redi[tid] = bi; redv2[tid] = fmaxf(b2, a1);
      } else {
        redv2[tid] = fmaxf(a2, b1);
      }
    }
  }
  __syncthreads();
  const float M    = redv[0];
  const int   gtok = redi[0];    // greedy token (argmax survives all masking)
  const float mx2  = redv2[0];   // second-largest value (top-2 never p-masked)
  __syncthreads();

  // key fetch for descent passes
  auto fetch_u = [&](int i) -> unsigned {
    if (USE_WS) return urow[i];
    return ford(xrow[i] / st);
  };

  // ---------------- top-k cutoff: 3-level radix descent on counts
  unsigned kcut = 0u;
  if (k > 0) {
    scan_detect(h, part, NB1, MODE_RANK, (float)k, false, &s_bin, &s_rem);
    int c1 = s_bin; float k2 = s_rem;
    __syncthreads();
    if (c1 < 0 || c1 >= NB1) c1 = 0;

    for (int b = tid; b < NB2; b += BLOCK) h[b] = 0.f;
    __syncthreads();
    for (int i = tid; i < v; i += BLOCK) {
      unsigned u = fetch_u(i);
      if ((int)(u >> 21) == c1) atomicAdd(&h[(u >> 10) & (NB2 - 1)], 1.0f);
    }
    __syncthreads();
    scan_detect(h, part, NB2, MODE_RANK, k2, false, &s_bin, &s_rem);
    int c2 = s_bin; float k3 = s_rem;
    __syncthreads();
    if (c2 < 0 || c2 >= NB2) c2 = 0;
    unsigned pfx = ((unsigned)c1 << 11) | (unsigned)c2;

    for (int b = tid; b < NB3; b += BLOCK) h[b] = 0.f;
    __syncthreads();
    for (int i = tid; i < v; i += BLOCK) {
      unsigned u = fetch_u(i);
      if ((u >> 10) == pfx) atomicAdd(&h[u & (NB3 - 1)], 1.0f);
    }
    __syncthreads();
    scan_detect(h, part, NB3, MODE_RANK, k3, false, &s_bin, &s_rem);
    int c3 = s_bin;
    __syncthreads();
    if (c3 < 0 || c3 >= NB3) c3 = 0;
    kcut = ((unsigned)c1 << 21) | ((unsigned)c2 << 10) | (unsigned)c3;
  }

  // ---------------- top-p cutoff: 3-level descent on exp-mass within {x>=kcut}
  for (int b = tid; b < NB1; b += BLOCK) h[b] = 0.f;
  __syncthreads();
  for (int i = tid; i < v; i += BLOCK) {
    unsigned u; float xv;
    if (USE_WS) { u = urow[i]; xv = iford(u); }
    else        { xv = xrow[i] / st; u = ford(xv); }
    if (u >= kcut) atomicAdd(&h[u >> 21], expf(xv - M));
  }
  __syncthreads();
  scan_detect(h, part, NB1, MODE_MASS, p, true, &s_bin, &s_rem);
  int d1 = s_bin; float R2 = s_rem;
  __syncthreads();
  if (d1 < 0 || d1 >= NB1) d1 = 0;

  for (int b = tid; b < NB2; b += BLOCK) h[b] = 0.f;
  __syncthreads();
  for (int i = tid; i < v; i += BLOCK) {
    unsigned u; float xv;
    if (USE_WS) { u = urow[i]; xv = iford(u); }
    else        { xv = xrow[i] / st; u = ford(xv); }
    if (u >= kcut && (int)(u >> 21) == d1)
      atomicAdd(&h[(u >> 10) & (NB2 - 1)], expf(xv - M));
  }
  __syncthreads();
  scan_detect(h, part, NB2, MODE_MASS, R2, false, &s_bin, &s_rem);
  int d2 = s_bin; float R3 = s_rem;
  __syncthreads();
  if (d2 < 0 || d2 >= NB2) d2 = 0;
  unsigned dpfx = ((unsigned)d1 << 11) | (unsigned)d2;

  for (int b = tid; b < NB3; b += BLOCK) h[b] = 0.f;
  __syncthreads();
  for (int i = tid; i < v; i += BLOCK) {
    unsigned u; float xv;
    if (USE_WS) { u = urow[i]; xv = iford(u); }
    else        { xv = xrow[i] / st; u = ford(xv); }
    if (u >= kcut && (u >> 10) == dpfx)
      atomicAdd(&h[u & (NB3 - 1)], expf(xv - M));
  }
  __syncthreads();
  scan_detect(h, part, NB3, MODE_MASS_FINAL, R3, false, &s_bin, &s_rem);
  int d3 = s_bin;
  __syncthreads();
  if (d3 < 0 || d3 >= NB3) d3 = 0;
  unsigned vstar = ((unsigned)d1 << 21) | ((unsigned)d2 << 10) | (unsigned)d3;

  unsigned m2u  = ford(mx2);
  unsigned pcut = vstar < m2u ? vstar : m2u;   // top-2 always kept
  unsigned cut  = kcut > pcut ? kcut : pcut;   // intersection of both masks

  // ---------------- Final pass: gumbel-max over kept set
  float bestv = -INFINITY; int besti = 0;
  for (int i = tid; i < v; i += BLOCK) {
    unsigned u; float xv;
    if (USE_WS) { u = urow[i]; xv = iford(u); }
    else        { xv = xrow[i] / st; u = ford(xv); }
    if (u >= cut) {
      float e = fmaxf(-logf(nrow[i]), 1e-10f);
      float g = xv - logf(e);                  // argmax(g) == argmax(probs/e)
      if (g > bestv) { bestv = g; besti = i; }
    }
  }
  __syncthreads();
  redv[tid] = bestv; redi[tid] = besti;
  for (int s = BLOCK / 2; s > 0; s >>= 1) {
    __syncthreads();
    if (tid < s) {
      float bv = redv[tid + s]; int bi = redi[tid + s];
      if (bv > redv[tid] || (bv == redv[tid] && bi < redi[tid])) {
        redv[tid] = bv; redi[tid] = bi;
      }
    }
  }
  __syncthreads();
  if (tid == 0) out[row] = (float)(greedy ? gtok : redi[0]);
}

extern "C" void kernel_launch(void* const* d_in, const int* in_sizes, int n_in,
                              void* d_out, int out_size, void* d_ws, size_t ws_size,
                              hipStream_t stream) {
  (void)n_in; (void)out_size;
  const float* logits = (const float*)d_in[0];
  const float* temps  = (const float*)d_in[1];
  const float* topps  = (const float*)d_in[2];
  const int*   topks  = (const int*)d_in[3];
  const float* noise  = (const float*)d_in[4];
  float* out = (float*)d_out;
  const int B = in_sizes[1];             // temperatures: one per row
  const int V = in_sizes[0] / B;         // vocab size
  const bool useWs = (d_ws != nullptr) &&
                     (ws_size >= (size_t)in_sizes[0] * sizeof(unsigned));
  if (useWs)
    sampler_kernel<true><<<dim3(B), dim3(BLOCK), 0, stream>>>(
        logits, temps, topps, topks, noise, out, (unsigned*)d_ws, V);
  else
    sampler_kernel<false><<<dim3(B), dim3(BLOCK), 0, stream>>>(
        logits, temps, topps, topks, noise, out, nullptr, V);
}